// RAMFeedForward_42606075576879
// MI455X (gfx1250) — compile-verified
//
#include <hip/hip_runtime.h>
#include <stdint.h>

// ---------------------------------------------------------------------------
// RAM feed-forward (weightless NN) for MI455X / gfx1250, wave32.
// B=16384, IN=1024, HID=4096, OUT=1024, 12 address bits, T=4096 entries.
//
// Bit-pack inputs/tables (32x smaller -> 4.5MB, L2-resident), gather bits
// from LDS (padded row layout, conflict-free), pack hidden state in LDS,
// coalesced NT output stores.  Packed-bits tile staged into LDS via the
// CDNA5 Tensor Data Mover (with LDS padding) + async global->LDS loads.
// ---------------------------------------------------------------------------

#define GLOBAL_AS __attribute__((address_space(1)))
#define LDS_AS    __attribute__((address_space(3)))

typedef uint32_t u32x4 __attribute__((ext_vector_type(4)));
typedef int      i32x4 __attribute__((ext_vector_type(4)));
typedef int      i32x8 __attribute__((ext_vector_type(8)));

#if __has_builtin(__builtin_amdgcn_tensor_load_to_lds) && \
    __has_builtin(__builtin_amdgcn_s_wait_tensorcnt)
#define HAVE_TDM 1
#else
#define HAVE_TDM 0
#endif

#if __has_builtin(__builtin_amdgcn_global_load_async_to_lds_b32) && \
    __has_builtin(__builtin_amdgcn_s_wait_asynccnt)
#define HAVE_ASYNC_LDS 1
#else
#define HAVE_ASYNC_LDS 0
#endif

#if HAVE_TDM
// DMA `rows` rows of 32 dwords from row-major global (stride 32 dwords) into
// LDS at byte offset lds_off, padding 1 dword after each 32 -> LDS row stride
// of 33 dwords (33 is odd -> conflict-free across the 64 LDS banks).
__device__ __forceinline__ void tdm_load_rows(const uint32_t* gsrc,
                                              uint32_t lds_off, int rows)
{
    uint64_t ga = (uint64_t)(uintptr_t)gsrc;
    u32x4 g0;
    g0[0] = 1u;                                   // count=1, user descriptor
    g0[1] = lds_off;                              // lds_addr (bytes)
    g0[2] = (uint32_t)ga;                         // global_addr[31:0]
    g0[3] = (uint32_t)(ga >> 32) | 0x80000000u;   // addr[56:32] | type=2<<30
    i32x8 g1;
    g1[0] = (int)((2u << 16) | (1u << 20) | (4u << 22)); // 4B elems, pad_en, 32dw interval, 1dw pad
    g1[1] = (int)(32u << 16);                     // tensor_dim0 = 32 (elems/row)
    g1[2] = (int)((uint32_t)rows << 16);          // tensor_dim1 = rows
    g1[3] = (int)(32u << 16);                     // tile_dim0 = 32
    g1[4] = rows;                                 // tile_dim1 = rows (tile_dim2=0)
    g1[5] = 32;                                   // tensor_dim0_stride = 32
    g1[6] = 0;
    g1[7] = 0;
    i32x4 gz = {0, 0, 0, 0};
#if __clang_major__ >= 23
    i32x8 gz8 = {0, 0, 0, 0, 0, 0, 0, 0};
    __builtin_amdgcn_tensor_load_to_lds(g0, g1, gz, gz, gz8, 0);
#else
    __builtin_amdgcn_tensor_load_to_lds(g0, g1, gz, gz, 0);
#endif
}
#endif

// Pack int32 {0,1} elements into u32 bit-words via wave32 ballot.
// NT loads: the 144MB of unpacked tables are read exactly once -- keep them
// out of L2 so the 4.5MB packed tables + output stream own it.
__global__ __launch_bounds__(256) void pack_bits_kernel(
    const int* __restrict__ src, uint32_t* __restrict__ dst, int nwords)
{
    const int lane = threadIdx.x & 31;
    const int wid  = (int)((blockIdx.x * blockDim.x + threadIdx.x) >> 5);
    const int nw   = (int)((gridDim.x * blockDim.x) >> 5);
    for (int w = wid; w < nwords; w += nw) {
        int v = __builtin_nontemporal_load(src + (size_t)w * 32 + lane);
        unsigned m = __builtin_amdgcn_ballot_w32((v & 1) != 0);
        if (lane == 0) dst[w] = m;
    }
}

// Main fused kernel, packed-table path.
// Block = 256 threads = 8 waves, owns 32 batch rows; lane = row-in-tile.
__global__ __launch_bounds__(256) void ram_ffn_packed(
    const uint32_t* __restrict__ bits_pk,   // [B,32]    packed input bits
    const int*      __restrict__ up_conn,   // [4096,12]
    const int*      __restrict__ gate_conn, // [4096,12]
    const int*      __restrict__ down_conn, // [1024,12]
    const uint32_t* __restrict__ up_pk,     // [4096,128] packed up table
    const uint32_t* __restrict__ gate_pk,   // [4096,128]
    const uint32_t* __restrict__ down_pk,   // [1024,128]
    float*          __restrict__ out)       // [B,1024]  0.0f/1.0f
{
    __shared__ uint32_t bitsT[32][33];   // [row][word], 33-dw stride: no conflicts
    __shared__ uint32_t hidT[128][32];   // [word][row] 16KB
    __shared__ uint32_t outT[32][32];    // [word][row]  4KB

    const int tid  = threadIdx.x;
    const int lane = tid & 31;
    const int wv   = tid >> 5;
    const int row0 = blockIdx.x << 5;

    // ---- Phase 0: stage packed input bits (32 rows x 128B) into LDS. ----
#if HAVE_TDM && HAVE_ASYNC_LDS
    if (wv == 0) {
        // Rows 0..15 via Tensor Data Mover (pads rows to 33-dword stride).
        tdm_load_rows(bits_pk + (size_t)row0 * 32,
                      (uint32_t)(uintptr_t)(LDS_AS uint32_t*)&bitsT[0][0], 16);
        __builtin_amdgcn_s_wait_tensorcnt(0);
    } else if (wv >= 4) {
        // Rows 16..31 via async global->LDS, 4 rows per wave, lane = word.
        int r0 = 16 + (wv - 4) * 4;
        for (int r = r0; r < r0 + 4; ++r) {
            GLOBAL_AS int* g = (GLOBAL_AS int*)
                (bits_pk + (size_t)(row0 + r) * 32 + lane);
            LDS_AS int* l = (LDS_AS int*)((LDS_AS uint32_t*)&bitsT[r][0] + lane);
            __builtin_amdgcn_global_load_async_to_lds_b32(g, l, 0, 0);
        }
        __builtin_amdgcn_s_wait_asynccnt(0);
    }
#elif HAVE_TDM
    if (wv == 0) {
        tdm_load_rows(bits_pk + (size_t)row0 * 32,
                      (uint32_t)(uintptr_t)(LDS_AS uint32_t*)&bitsT[0][0], 32);
        __builtin_amdgcn_s_wait_tensorcnt(0);
    }
#elif HAVE_ASYNC_LDS
    {
        int r0 = wv * 4;
        for (int r = r0; r < r0 + 4; ++r) {
            GLOBAL_AS int* g = (GLOBAL_AS int*)
                (bits_pk + (size_t)(row0 + r) * 32 + lane);
            LDS_AS int* l = (LDS_AS int*)((LDS_AS uint32_t*)&bitsT[r][0] + lane);
            __builtin_amdgcn_global_load_async_to_lds_b32(g, l, 0, 0);
        }
        __builtin_amdgcn_s_wait_asynccnt(0);
    }
#else
    for (int i = tid; i < 1024; i += 256) {
        int r = i >> 5, w = i & 31;
        bitsT[r][w] = bits_pk[(size_t)(row0 + r) * 32 + w];
    }
#endif
    __syncthreads();

    // ---- Phase 1: hidden = up(x) & gate(x), packed into hidT. ----
    const int nEnd = wv * 512 + 512;
    for (int n0 = wv * 512; n0 < nEnd; n0 += 32) {
        if (n0 + 32 < nEnd) {           // warm next conn rows (global_prefetch_b8)
            __builtin_prefetch(up_conn   + (size_t)(n0 + 32) * 12, 0, 0);
            __builtin_prefetch(gate_conn + (size_t)(n0 + 32) * 12, 0, 0);
        }
        uint32_t word = 0;
        for (int j = 0; j < 32; ++j) {
            const int nn = n0 + j;
            const int* cu = up_conn   + (size_t)nn * 12;   // wave-uniform -> s_load
            const int* cg = gate_conn + (size_t)nn * 12;
            int au = 0, ag = 0;
#pragma unroll
            for (int k = 0; k < 12; ++k) {
                int iu = cu[k];
                au |= (int)((bitsT[lane][iu >> 5] >> (iu & 31)) & 1u) << k;
                int ig = cg[k];
                ag |= (int)((bitsT[lane][ig >> 5] >> (ig & 31)) & 1u) << k;
            }
            // Per-lane 4B gather inside this neuron's 512B row window.
            uint32_t ub = (up_pk  [(size_t)nn * 128 + (au >> 5)] >> (au & 31)) & 1u;
            uint32_t gb = (gate_pk[(size_t)nn * 128 + (ag >> 5)] >> (ag & 31)) & 1u;
            word |= (ub & gb) << j;
        }
        hidT[n0 >> 5][lane] = word;
    }
    __syncthreads();

    // ---- Phase 2: out = down(hidden) ^ bits, packed into outT. ----
    const int oEnd = wv * 128 + 128;
    for (int o0 = wv * 128; o0 < oEnd; o0 += 32) {
        if (o0 + 32 < oEnd)
            __builtin_prefetch(down_conn + (size_t)(o0 + 32) * 12, 0, 0);
        uint32_t word = 0;
        for (int j = 0; j < 32; ++j) {
            const int oo = o0 + j;
            const int* cd = down_conn + (size_t)oo * 12;
            int a = 0;
#pragma unroll
            for (int k = 0; k < 12; ++k) {
                int ih = cd[k];
                a |= (int)((hidT[ih >> 5][lane] >> (ih & 31)) & 1u) << k;
            }
            word |= ((down_pk[(size_t)oo * 128 + (a >> 5)] >> (a & 31)) & 1u) << j;
        }
        outT[o0 >> 5][lane] = word ^ bitsT[lane][o0 >> 5];   // residual XOR
    }
    __syncthreads();

    // ---- Phase 3: expand packed output to float, coalesced NT stores. ----
    for (int i = tid; i < 32 * 1024; i += 256) {
        int r = i >> 10, o = i & 1023;
        float v = (float)((outT[o >> 5][r] >> (o & 31)) & 1u);
        __builtin_nontemporal_store(v, out + (size_t)(row0 + r) * 1024 + o);
    }
}

// Fallback path (workspace too small): ballot-pack bits in-kernel, gather
// directly from the int32 tables (144MB -- still fits the 192MB L2).
__global__ __launch_bounds__(256) void ram_ffn_direct(
    const int* __restrict__ bits,        // [B,1024]
    const int* __restrict__ up_conn,
    const int* __restrict__ gate_conn,
    const int* __restrict__ down_conn,
    const int* __restrict__ up_table,    // [4096,4096]
    const int* __restrict__ gate_table,  // [4096,4096]
    const int* __restrict__ down_table,  // [1024,4096]
    float*     __restrict__ out)
{
    __shared__ uint32_t bitsT[32][33];
    __shared__ uint32_t hidT[128][32];
    __shared__ uint32_t outT[32][32];

    const int tid  = threadIdx.x;
    const int lane = tid & 31;
    const int wv   = tid >> 5;
    const int row0 = blockIdx.x << 5;

    // Ballot-pack raw input bits. 1024 (r,w) pairs, 128 per wave.
    for (int i = 0; i < 128; ++i) {
        int p = wv * 128 + i;
        int r = p >> 5, w = p & 31;
        int v = bits[(size_t)(row0 + r) * 1024 + w * 32 + lane];
        unsigned m = __builtin_amdgcn_ballot_w32((v & 1) != 0);
        if (lane == 0) bitsT[r][w] = m;
    }
    __syncthreads();

    const int nEnd = wv * 512 + 512;
    for (int n0 = wv * 512; n0 < nEnd; n0 += 32) {
        uint32_t word = 0;
        for (int j = 0; j < 32; ++j) {
            const int nn = n0 + j;
            const int* cu = up_conn   + (size_t)nn * 12;
            const int* cg = gate_conn + (size_t)nn * 12;
            int au = 0, ag = 0;
#pragma unroll
            for (int k = 0; k < 12; ++k) {
                int iu = cu[k];
                au |= (int)((bitsT[lane][iu >> 5] >> (iu & 31)) & 1u) << k;
                int ig = cg[k];
                ag |= (int)((bitsT[lane][ig >> 5] >> (ig & 31)) & 1u) << k;
            }
            uint32_t ub = (uint32_t)up_table  [(size_t)nn * 4096 + au] & 1u;
            uint32_t gb = (uint32_t)gate_table[(size_t)nn * 4096 + ag] & 1u;
            word |= (ub & gb) << j;
        }
        hidT[n0 >> 5][lane] = word;
    }
    __syncthreads();

    const int oEnd = wv * 128 + 128;
    for (int o0 = wv * 128; o0 < oEnd; o0 += 32) {
        uint32_t word = 0;
        for (int j = 0; j < 32; ++j) {
            const int oo = o0 + j;
            const int* cd = down_conn + (size_t)oo * 12;
            int a = 0;
#pragma unroll
            for (int k = 0; k < 12; ++k) {
                int ih = cd[k];
                a |= (int)((hidT[ih >> 5][lane] >> (ih & 31)) & 1u) << k;
            }
            word |= ((uint32_t)down_table[(size_t)oo * 4096 + a] & 1u) << j;
        }
        outT[o0 >> 5][lane] = word ^ bitsT[lane][o0 >> 5];
    }
    __syncthreads();

    for (int i = tid; i < 32 * 1024; i += 256) {
        int r = i >> 10, o = i & 1023;
        float v = (float)((outT[o >> 5][r] >> (o & 31)) & 1u);
        __builtin_nontemporal_store(v, out + (size_t)(row0 + r) * 1024 + o);
    }
}

extern "C" void kernel_launch(void* const* d_in, const int* in_sizes, int n_in,
                              void* d_out, int out_size, void* d_ws, size_t ws_size,
                              hipStream_t stream)
{
    const int* bits       = (const int*)d_in[0];
    const int* up_conn    = (const int*)d_in[1];
    const int* gate_conn  = (const int*)d_in[2];
    const int* down_conn  = (const int*)d_in[3];
    const int* up_table   = (const int*)d_in[4];
    const int* gate_table = (const int*)d_in[5];
    const int* down_table = (const int*)d_in[6];
    float* out = (float*)d_out;

    const int B = in_sizes[0] / 1024;              // 16384
    const size_t UPW  = (size_t)4096 * 128;        // packed words per big table
    const size_t DNW  = (size_t)1024 * 128;
    const size_t BITW = (size_t)B * 32;
    const size_t need = (UPW * 2 + DNW + BITW) * sizeof(uint32_t);  // ~6.5MB

    const int nblocks = B / 32;                    // 512 blocks x 8 waves

    if (ws_size >= need) {
        uint32_t* up_pk = (uint32_t*)d_ws;
        uint32_t* gt_pk = up_pk + UPW;
        uint32_t* dn_pk = gt_pk + UPW;
        uint32_t* bt_pk = dn_pk + DNW;
        pack_bits_kernel<<<512, 256, 0, stream>>>(up_table,   up_pk, (int)UPW);
        pack_bits_kernel<<<512, 256, 0, stream>>>(gate_table, gt_pk, (int)UPW);
        pack_bits_kernel<<<128, 256, 0, stream>>>(down_table, dn_pk, (int)DNW);
        pack_bits_kernel<<<512, 256, 0, stream>>>(bits,       bt_pk, (int)BITW);
        ram_ffn_packed<<<nblocks, 256, 0, stream>>>(
            bt_pk, up_conn, gate_conn, down_conn, up_pk, gt_pk, dn_pk, out);
    } else {
        ram_ffn_direct<<<nblocks, 256, 0, stream>>>(
            bits, up_conn, gate_conn, down_conn,
            up_table, gate_table, down_table, out);
    }
}